// TuckerFacLayer_45294725104082
// MI455X (gfx1250) — compile-verified
//
#include <hip/hip_runtime.h>
#include <hip/hip_bf16.h>

// ---------------------------------------------------------------------------
// Tucker-factorized layer:
//   out[b,v,t,n,d,e] = sum_a x[b,v,t,n,d,a] * M_bv[a,e]
//   M_bv[a,e] = sum_{A,i,j} fv[b,v,A] core[A,i,j] fac_in[i,a] fac_out[j,e]
//
// Roofline: 8.6 GFLOP vs 537 MB traffic = 16 FLOP/B -> memory-bound at
// 23.3 TB/s (~23 us floor). Keep fp32; use V_WMMA_F32_16X16X4_F32.
// Cache policy: x/out streams (537 MB) > L2 (192 MB) -> non-temporal;
// Mt (256 KB, reused by every wave) stays regular-temporal in L2.
// ---------------------------------------------------------------------------

#define B_ 2
#define V_ 8
#define T_ 4
#define N_ 1024
#define D_ 16
#define FIN_ 64
#define FOUT_ 64
#define R_ 16
#define RV_ 4

#define BV_ (B_ * V_)                 // 16
#define ROWS_PER_BV_ (T_ * N_ * D_)   // 65536 rows of length 64

typedef __attribute__((ext_vector_type(2))) float v2f;
typedef __attribute__((ext_vector_type(8))) float v8f;

// ---------------------------------------------------------------------------
// Stage 1: build the 16 fused 64x64 matrices, stored TRANSPOSED:
//   Mt[bv][e][a] = M_bv[a][e]
// so that stage-2 B-fragment loads are contiguous float2 per lane.
// ---------------------------------------------------------------------------
__global__ __launch_bounds__(64) void tucker_build_m_kernel(
    const int* __restrict__ var_idx,      // (B,V)
    const float* __restrict__ core,       // (RV, R, R)
    const float* __restrict__ factor_vars,// (NVAR, RV)
    const float* __restrict__ fac_in,     // (R, FIN)
    const float* __restrict__ fac_out,    // (R, FOUT)
    float* __restrict__ Mt)               // (BV, FOUT, FIN)
{
    __shared__ float fv[RV_];
    __shared__ float G[R_][R_];      // sum_A fv[A] * core[A]
    __shared__ float H[R_][FOUT_];   // G @ fac_out

    const int bv = blockIdx.x;
    const int t  = threadIdx.x;      // 0..63

    if (t < RV_) {
        const int vid = var_idx[bv];
        fv[t] = factor_vars[vid * RV_ + t];
    }
    __syncthreads();

    // G[i][j] = sum_A fv[A] * core[A][i][j]   (256 elems / 64 threads)
    for (int idx = t; idx < R_ * R_; idx += 64) {
        float s = 0.f;
        #pragma unroll
        for (int A = 0; A < RV_; ++A) s += fv[A] * core[A * R_ * R_ + idx];
        G[idx >> 4][idx & 15] = s;
    }
    __syncthreads();

    // H[i][e] = sum_j G[i][j] * fac_out[j][e]   (1024 elems / 64 threads)
    for (int idx = t; idx < R_ * FOUT_; idx += 64) {
        const int i = idx >> 6, e = idx & 63;
        float s = 0.f;
        #pragma unroll
        for (int j = 0; j < R_; ++j) s += G[i][j] * fac_out[j * FOUT_ + e];
        H[i][e] = s;
    }
    __syncthreads();

    // Mt[e][a] = sum_i fac_in[i][a] * H[i][e]   (4096 elems / 64 threads)
    float* Mb = Mt + (size_t)bv * FOUT_ * FIN_;
    for (int idx = t; idx < FOUT_ * FIN_; idx += 64) {
        const int e = idx >> 6, a = idx & 63;
        float s = 0.f;
        #pragma unroll
        for (int i = 0; i < R_; ++i) s += fac_in[i * FIN_ + a] * H[i][e];
        Mb[idx] = s;
    }
}

// ---------------------------------------------------------------------------
// Stage 2: streaming GEMM.  Per (b,v): X (65536 x 64) @ M (64 x 64).
// One wave32 computes a 16-row x 64-col output tile via 4x16 f32 WMMAs.
//
// A fragment (16x4 f32): lane L (row = L%16):
//   lanes 0-15 : V0 = K=k0+0, V1 = K=k0+1
//   lanes 16-31: V0 = K=k0+2, V1 = K=k0+3
// -> per lane: one float2 at X[row][k0 + 2*half].      (non-temporal)
//
// B fragment (4x16 f32), mirrored striping: lane L (col e = c*16 + L%16):
//   V0 = M[k0+2*half][e], V1 = M[k0+2*half+1][e]
// -> contiguous float2 from Mt[e][k0 + 2*half].        (regular, L2-hot)
//
// C/D (16x16 f32): lane L, VGPR i -> row = i + 8*(L/16), col = L%16.
// Output stores are non-temporal (write-once stream).
// ---------------------------------------------------------------------------
__global__ __launch_bounds__(256) void tucker_wmma_gemm_kernel(
    const float* __restrict__ x,    // (BV, 65536, 64)
    const float* __restrict__ Mt,   // (BV, 64, 64) = M transposed [e][a]
    float* __restrict__ out)        // (BV, 65536, 64)
{
    const int wave = threadIdx.x >> 5;
    const int lane = threadIdx.x & 31;
    const int half = lane >> 4;     // 0: lanes 0-15, 1: lanes 16-31
    const int l16  = lane & 15;

    const long tile    = (long)blockIdx.x * 8 + wave;   // 16-row output tile
    const long rowbase = tile * 16;                     // global row
    const int  bv      = (int)(rowbase / ROWS_PER_BV_);
    const long lrow    = rowbase % ROWS_PER_BV_;        // row within slab

    const float* __restrict__ Xb = x   + (size_t)bv * ROWS_PER_BV_ * FIN_;
    const float* __restrict__ Mb = Mt  + (size_t)bv * FOUT_ * FIN_;
    float*       __restrict__ Ob = out + (size_t)bv * ROWS_PER_BV_ * FOUT_;

    // ---- load all A fragments for this 16-row tile (K = 0..63), NT ----
    v2f a[16];
    const float* arow = Xb + (size_t)(lrow + l16) * FIN_;
    #pragma unroll
    for (int k = 0; k < 16; ++k)
        a[k] = __builtin_nontemporal_load((const v2f*)(arow + 4 * k + 2 * half));

    // ---- 4 column tiles of 16 output columns each ----
    #pragma unroll
    for (int c = 0; c < 4; ++c) {
        v2f bf[16];
        const float* brow = Mb + (size_t)(c * 16 + l16) * FIN_;  // Mt[e][*]
        #pragma unroll
        for (int k = 0; k < 16; ++k)
            bf[k] = *(const v2f*)(brow + 4 * k + 2 * half);

        v8f acc = {};
        #pragma unroll
        for (int k = 0; k < 16; ++k)
            acc = __builtin_amdgcn_wmma_f32_16x16x4_f32(
                /*neg_a=*/false, a[k], /*neg_b=*/false, bf[k],
                /*c_mod=*/(short)0, acc, /*reuse_a=*/false, /*reuse_b=*/false);

        float* obase = Ob + (size_t)lrow * FOUT_ + c * 16 + l16;
        #pragma unroll
        for (int i = 0; i < 8; ++i) {
            const int row = i + 8 * half;
            __builtin_nontemporal_store(acc[i], obase + (size_t)row * FOUT_);
        }
    }
}

extern "C" void kernel_launch(void* const* d_in, const int* in_sizes, int n_in,
                              void* d_out, int out_size, void* d_ws, size_t ws_size,
                              hipStream_t stream) {
    (void)in_sizes; (void)n_in; (void)out_size; (void)ws_size;

    const float* x           = (const float*)d_in[0];
    const int*   var_idx     = (const int*)d_in[1];
    const float* core        = (const float*)d_in[2];
    const float* factor_vars = (const float*)d_in[3];
    const float* fac_in      = (const float*)d_in[4];
    const float* fac_out     = (const float*)d_in[5];
    float*       out         = (float*)d_out;
    float*       Mt          = (float*)d_ws;   // 16 * 64 * 64 * 4B = 256 KB

    // Stage 1: tiny setup (fuse core/fv/fac_in/fac_out into 16 64x64 mats)
    tucker_build_m_kernel<<<BV_, 64, 0, stream>>>(
        var_idx, core, factor_vars, fac_in, fac_out, Mt);

    // Stage 2: streaming WMMA GEMM over 1,048,576 rows.
    // tiles = (16 * 65536) / 16 = 65536 ; 8 waves/block -> 8192 blocks
    tucker_wmma_gemm_kernel<<<8192, 256, 0, stream>>>(x, Mt, out);
}